// MessagePassing_38474317037707
// MI455X (gfx1250) — compile-verified
//
#include <hip/hip_runtime.h>

typedef __attribute__((ext_vector_type(2))) float v2f;
typedef __attribute__((ext_vector_type(4))) float v4f;
typedef __attribute__((ext_vector_type(8))) float v8f;

#define DFEAT 64   // input feature dim (K of the precompute GEMM)
#define DOUT  64   // output dim
#define YCOLS 128  // Y row = [x@W_top (64) | x@W_bot (64)]

// ---------------------------------------------------------------------------
// Kernel 1: Y[n, 0:64]  = X[n,:] @ W[0:64, :]
//           Y[n,64:128] = X[n,:] @ W[64:128, :]
// One block = 16 node rows; 8 waves, each wave owns a 16-wide column slice.
// WMMA f32 16x16x4, K-loop of 16 steps.
// ---------------------------------------------------------------------------
__global__ __launch_bounds__(256) void gnn_precompute_gemm(
    const float* __restrict__ X, const float* __restrict__ W,
    float* __restrict__ Y, int n_nodes)
{
    const int lane  = threadIdx.x & 31;
    const int wave  = threadIdx.x >> 5;        // 0..7 -> column slice of Y
    const int m0    = blockIdx.x << 4;         // 16 nodes per block
    const int mrow  = lane & 15;
    const int khalf = lane >> 4;               // 0: K={k,k+1}; 1: K={k+2,k+3}

    // Column this lane supplies for B (and writes for D).
    const int j     = (wave << 4) + mrow;      // 0..127
    const int jc    = j & 63;                  // column within W
    const int jkoff = (j >> 6) << 6;           // +64 W-row offset for cols >= 64

    int arow = m0 + mrow;
    if (arow >= n_nodes) arow = n_nodes - 1;   // 50000 % 16 == 0; safety only
    const float* __restrict__ Arow = X + (size_t)arow * DFEAT;

    v8f acc = (v8f){0.f, 0.f, 0.f, 0.f, 0.f, 0.f, 0.f, 0.f};

#pragma unroll
    for (int kk = 0; kk < DFEAT; kk += 4) {
        const int ka = kk + (khalf << 1);      // this lane's two K indices
        // A tile 16x4: contiguous pair -> single b64 load (ka is even, 8B aligned)
        v2f a = *(const v2f*)(Arow + ka);
        // B tile 4x16: W rows are 64 floats apart -> two b32 loads (L1/L2 hits)
        v2f bm;
        bm.x = W[(size_t)(ka + jkoff) * DOUT + jc];
        bm.y = W[(size_t)(ka + 1 + jkoff) * DOUT + jc];
        acc = __builtin_amdgcn_wmma_f32_16x16x4_f32(
            /*neg_a=*/false, a, /*neg_b=*/false, bm,
            /*c_mod=*/(short)0, acc, /*reuse_a=*/false, /*reuse_b=*/false);
    }

    // D layout: VGPR r holds M = r + 8*khalf, N = lane&15 (== mrow).
#pragma unroll
    for (int r = 0; r < 8; ++r) {
        int m = m0 + r + (khalf << 3);
        if (m >= n_nodes) m = n_nodes - 1;     // benign duplicate in tail case
        Y[(size_t)m * YCOLS + j] = acc[r];
    }
}

// ---------------------------------------------------------------------------
// Kernel 2: out[e, :] = Y[src[e], 0:64] + Y[dst[e], 64:128] + b
// 16 threads per edge, float4 per thread. Y (25.6 MB) is L2-resident on
// MI455X (192 MB L2) so the random gathers hit L2; the 205 MB output stream
// uses non-temporal stores to avoid evicting it.
// ---------------------------------------------------------------------------
__global__ __launch_bounds__(256) void gnn_edge_combine(
    const v4f* __restrict__ Y, const int* __restrict__ src,
    const int* __restrict__ dst, const v4f* __restrict__ bias,
    v4f* __restrict__ out, int n_edges)
{
    const int tid = blockIdx.x * blockDim.x + threadIdx.x;
    const int e = tid >> 4;                    // edge
    const int q = tid & 15;                    // float4 index within 64 outputs
    if (e >= n_edges) return;

    const int s = src[e];
    const int d = dst[e];

    v4f vs = Y[(size_t)s * (YCOLS / 4) + q];            // Y[s, 0:64] slice
    v4f vd = Y[(size_t)d * (YCOLS / 4) + 16 + q];       // Y[d, 64:128] slice
    v4f vb = bias[q];
    v4f r = vs + vd + vb;

    __builtin_nontemporal_store(r, &out[(size_t)e * (DOUT / 4) + q]);
}

// ---------------------------------------------------------------------------
// Inputs (setup_inputs order): 0=x [N,64] f32, 1=W [128,64] f32, 2=b [64] f32,
//                              3=src [E] i32, 4=dst [E] i32
// ---------------------------------------------------------------------------
extern "C" void kernel_launch(void* const* d_in, const int* in_sizes, int n_in,
                              void* d_out, int out_size, void* d_ws, size_t ws_size,
                              hipStream_t stream) {
    const float* X   = (const float*)d_in[0];
    const float* W   = (const float*)d_in[1];
    const float* b   = (const float*)d_in[2];
    const int*   src = (const int*)d_in[3];
    const int*   dst = (const int*)d_in[4];
    float*       out = (float*)d_out;

    const int n_nodes = in_sizes[0] / DFEAT;
    const int n_edges = in_sizes[3];

    float* Y = (float*)d_ws;   // n_nodes * 128 floats = 25.6 MB scratch

    const int node_tiles = (n_nodes + 15) / 16;
    gnn_precompute_gemm<<<node_tiles, 256, 0, stream>>>(X, W, Y, n_nodes);

    const long total_threads = (long)n_edges * 16;
    const int  blocks = (int)((total_threads + 255) / 256);
    gnn_edge_combine<<<blocks, 256, 0, stream>>>(
        (const v4f*)Y, src, dst, (const v4f*)b, (v4f*)out, n_edges);
}